// ContactModel_68650757260066
// MI455X (gfx1250) — compile-verified
//
#include <hip/hip_runtime.h>

#if defined(__has_builtin)
#  if __has_builtin(__builtin_amdgcn_global_store_async_from_lds_b32)
#    define CK_ASYNC_STORE 1
#  endif
#  if __has_builtin(__builtin_amdgcn_s_wait_asynccnt)
#    define CK_WAIT_ASYNC 1
#  endif
#endif

namespace {

constexpr int THREADS = 256;

#if defined(CK_ASYNC_STORE)
typedef __attribute__((address_space(1))) int GlobalInt;   // HIP device-global
typedef __attribute__((address_space(3))) int SharedInt;   // LDS
#endif

// ---- load exactly the needed fields of one batch row --------------------
// joints row: 72 floats (24 bodies x 3): need floats 12..17, 27..32.
// joints_ori row: 216 floats (24 x 9): need floats 36..53, 81..98.
// Row strides (288B / 864B) are multiples of 16 -> float4 loads aligned.
__device__ __forceinline__ void load_row(const float* __restrict__ jr,
                                         const float* __restrict__ orw,
                                         float* __restrict__ j,   // [12]
                                         float* __restrict__ R)   // [36]
{
    const float4 a0 = *reinterpret_cast<const float4*>(jr + 12); // 12..15
    const float2 a1 = *reinterpret_cast<const float2*>(jr + 16); // 16,17
    const float4 a2 = *reinterpret_cast<const float4*>(jr + 24); // 24..27
    const float4 a3 = *reinterpret_cast<const float4*>(jr + 28); // 28..31
    const float  a4 = jr[32];
    j[0]=a0.x;  j[1]=a0.y;  j[2]=a0.z;   // body 4
    j[3]=a0.w;  j[4]=a1.x;  j[5]=a1.y;   // body 5
    j[6]=a2.w;  j[7]=a3.x;  j[8]=a3.y;   // body 9
    j[9]=a3.z;  j[10]=a3.w; j[11]=a4;    // body 10

    const float4 r0 = *reinterpret_cast<const float4*>(orw + 36);
    const float4 r1 = *reinterpret_cast<const float4*>(orw + 40);
    const float4 r2 = *reinterpret_cast<const float4*>(orw + 44);
    const float4 r3 = *reinterpret_cast<const float4*>(orw + 48);
    const float2 r4 = *reinterpret_cast<const float2*>(orw + 52);
    const float4 r5 = *reinterpret_cast<const float4*>(orw + 80);
    const float4 r6 = *reinterpret_cast<const float4*>(orw + 84);
    const float4 r7 = *reinterpret_cast<const float4*>(orw + 88);
    const float4 r8 = *reinterpret_cast<const float4*>(orw + 92);
    const float4 r9 = *reinterpret_cast<const float4*>(orw + 96);
    // body 4: floats 36..44
    R[0]=r0.x; R[1]=r0.y; R[2]=r0.z; R[3]=r0.w;
    R[4]=r1.x; R[5]=r1.y; R[6]=r1.z; R[7]=r1.w;
    R[8]=r2.x;
    // body 5: floats 45..53
    R[9]=r2.y;  R[10]=r2.z; R[11]=r2.w;
    R[12]=r3.x; R[13]=r3.y; R[14]=r3.z; R[15]=r3.w;
    R[16]=r4.x; R[17]=r4.y;
    // body 9: floats 81..89
    R[18]=r5.y; R[19]=r5.z; R[20]=r5.w;
    R[21]=r6.x; R[22]=r6.y; R[23]=r6.z; R[24]=r6.w;
    R[25]=r7.x; R[26]=r7.y;
    // body 10: floats 90..98
    R[27]=r7.z; R[28]=r7.w;
    R[29]=r8.x; R[30]=r8.y; R[31]=r8.z; R[32]=r8.w;
    R[33]=r9.x; R[34]=r9.y; R[35]=r9.z;
}

// p[s*3+c] = joints[body][c] + (R[body] * local[s])[c]
__device__ __forceinline__ void pos_from(const float* __restrict__ j,
                                         const float* __restrict__ R,
                                         float* __restrict__ p)   // [36]
{
    constexpr int BS[12] = {0,0,0,0,1,1,2,2,2,2,3,3};
    constexpr float LX[12] = {
        0.00190115788407966f, 0.148386399942063f, 0.133001170607051f,
        0.0662346661991635f,  0.06f,              0.045f,
        0.00190115788407966f, 0.148386399942063f, 0.133001170607051f,
        0.0662346661991635f,  0.06f,              0.045f };
    constexpr float LZ[12] = {
        -0.00382630379623308f, -0.028713422052654f,  0.0516362473449566f,
         0.0263641606741698f,  -0.0187603084619177f, 0.0618569567549652f,
         0.00382630379623308f,  0.028713422052654f, -0.0516362473449566f,
        -0.0263641606741698f,   0.0187603084619177f,-0.0618569567549652f };
    constexpr float LY = -0.01f;
#pragma unroll
    for (int s = 0; s < 12; ++s) {
        const int k9 = BS[s]*9, k3 = BS[s]*3;
        const float lx = LX[s], lz = LZ[s];
        p[s*3+0] = j[k3+0] + R[k9+0]*lx + R[k9+1]*LY + R[k9+2]*lz;
        p[s*3+1] = j[k3+1] + R[k9+3]*lx + R[k9+4]*LY + R[k9+5]*lz;
        p[s*3+2] = j[k3+2] + R[k9+6]*lx + R[k9+7]*LY + R[k9+8]*lz;
    }
}

__global__ void __launch_bounds__(THREADS)
contact_kernel(const float* __restrict__ joints,
               const float* __restrict__ ori,
               float* __restrict__ out, int B)
{
    // phase 1: pos sharing (256 rows x 36 floats); phase 2 (reused): 27-scalar
    // output transpose staging (256 x 28).
    __shared__ float sm[THREADS * 36];

    const int tid = threadIdx.x;
    const int b = blockIdx.x * THREADS + tid;
    const bool active = (b < B);
    const size_t sB = (size_t)B;

    float p[36];   // this row's sphere positions
    float qv[36];  // previous row's sphere positions

    if (active) {
        // only lane 0 of the block must fetch the b0-1 row from memory;
        // everyone else gets pos[b-1] from a neighbor through LDS.
        if (tid == 0) {
            const size_t bp = (b > 0) ? (size_t)(b - 1) : 0;
            const float* jrp = joints + bp * 72;
            const float* orp = ori    + bp * 216;
            // gfx1250 global_prefetch_b8: overlap this lone-lane gather
            __builtin_prefetch(jrp + 12, 0, 3);
            __builtin_prefetch(orp + 36, 0, 3);
            __builtin_prefetch(orp + 80, 0, 3);
            float jq[12], Rq[36];
            load_row(jrp, orp, jq, Rq);
            pos_from(jq, Rq, qv);
        }
        const size_t bb = (size_t)b;
        float jc[12], Rc[36];
        load_row(joints + bb * 72, ori + bb * 216, jc, Rc);
        pos_from(jc, Rc, p);
        float4* pv = reinterpret_cast<float4*>(&sm[tid * 36]);
#pragma unroll
        for (int q = 0; q < 9; ++q)
            pv[q] = make_float4(p[4*q+0], p[4*q+1], p[4*q+2], p[4*q+3]);
    }
    __syncthreads();

    float res[27];
#pragma unroll
    for (int i = 0; i < 27; ++i) res[i] = 0.0f;

    if (active) {
        if (tid != 0) {
            const float4* sv = reinterpret_cast<const float4*>(&sm[(tid-1) * 36]);
#pragma unroll
            for (int q = 0; q < 9; ++q) {
                const float4 v = sv[q];
                qv[4*q+0]=v.x; qv[4*q+1]=v.y; qv[4*q+2]=v.z; qv[4*q+3]=v.w;
            }
        }

        // accumulators: [0]=right (s 0..5), [1]=left (s 6..11)
        float fsx[2]={0,0}, fsy[2]={0,0}, fsz[2]={0,0};
        float wsum[2]={0,0}, xw[2]={0,0}, zw[2]={0,0};
        float fmx[2]={0,0}, fmy[2]={0,0}, fmz[2]={0,0};
        float crx[2]={0,0}, cry[2]={0,0}, crz[2]={0,0};
        float fl[36];

        // 4/3 * k * sqrt(RADIUS*k), k = 0.5*STIFFNESS^(2/3)
        constexpr float FH_COEF = 8432.74043f;

#pragma unroll
        for (int s = 0; s < 12; ++s) {
            const int g = (s < 6) ? 0 : 1;
            const float cx = p[s*3+0], cy = p[s*3+1], cz = p[s*3+2];
            const float vx = (cx - qv[s*3+0]) * 20.0f;   // 1/DT
            const float vy = (cy - qv[s*3+1]) * 20.0f;
            const float vz = (cz - qv[s*3+2]) * 20.0f;

            const float ind = -cy;                        // GROUND_HEIGHT=0
            const float iv  = -vy;
            const float xq  = ind * ind + 1e-5f;
            const float fH  = FH_COEF * sqrtf(xq * sqrtf(xq));  // xq^0.75
            const float fHd = fH * (1.0f + 0.3f * iv);          // 1.5*DISS
            // 0.5*tanh(z)+0.5 == sigmoid(2z), overflow-safe via v_exp_f32
            const float s1 = 1.0f/(1.0f + __expf(-100.0f*(iv + 3.3333333f))) + 1e-16f;
            const float s2 = 1.0f/(1.0f + __expf(-600.0f*ind)) + 1e-16f;
            const float fn = s1 * s2 * fHd;

            const float vslip = sqrtf(vx*vx + vz*vz + 1e-5f);
            const float vrel  = vslip * 5.0f;            // /TRANSITION_VEL
            // STATIC==DYNAMIC -> mu term constant 0.8
            const float ffr   = fn * (fminf(vrel, 1.0f)*0.8f + 0.5f*vslip);
            const float scale = ffr / (vslip + 1e-5f);
            const float fx = -scale * vx;
            const float fy = fn;
            const float fz = -scale * vz;

            fl[s*3+0]=fx; fl[s*3+1]=fy; fl[s*3+2]=fz;

            fsx[g]+=fx; fsy[g]+=fy; fsz[g]+=fz;
            const bool valid = fy > 0.0f;
            const float w  = valid ? fy : 0.0f;
            const float mx = valid ? fx : 0.0f;
            const float mz = valid ? fz : 0.0f;
            wsum[g]+=w; xw[g]+=cx*w; zw[g]+=cz*w;
            fmx[g]+=mx; fmy[g]+=w; fmz[g]+=mz;
            crx[g] += cy*mz - cz*w;
            cry[g] += cz*mx - cx*mz;
            crz[g] += cx*w  - cy*mx;
        }

        // torque = sum cross(p, fm) - cross(cop, sum fm); cop.y == 0
        auto fin = [](float W, float XW, float ZW,
                      float FX, float FY, float FZ,
                      float CX, float CY, float CZ,
                      float& ocx, float& ocz,
                      float& otx, float& oty, float& otz) {
            const bool  has = W > 0.0f;
            const float tws = has ? W : 1.0f;
            ocx = has ? XW / tws : 0.0f;
            ocz = has ? ZW / tws : 0.0f;
            otx = CX + ocz * FY;
            oty = CY - (ocz * FX - ocx * FZ);
            otz = CZ - ocx * FY;
        };

        float copx[3], copz[3], tqx[3], tqy[3], tqz[3]; // 0=all,1=right,2=left
        fin(wsum[0]+wsum[1], xw[0]+xw[1], zw[0]+zw[1],
            fmx[0]+fmx[1], fmy[0]+fmy[1], fmz[0]+fmz[1],
            crx[0]+crx[1], cry[0]+cry[1], crz[0]+crz[1],
            copx[0], copz[0], tqx[0], tqy[0], tqz[0]);
        fin(wsum[0], xw[0], zw[0], fmx[0], fmy[0], fmz[0],
            crx[0], cry[0], crz[0],
            copx[1], copz[1], tqx[1], tqy[1], tqz[1]);
        fin(wsum[1], xw[1], zw[1], fmx[1], fmy[1], fmz[1],
            crx[1], cry[1], crz[1],
            copx[2], copz[2], tqx[2], tqy[2], tqz[2]);

        res[0]=fsx[0]+fsx[1]; res[1]=fsy[0]+fsy[1]; res[2]=fsz[0]+fsz[1];
        res[3]=tqx[0];  res[4]=tqy[0];  res[5]=tqz[0];
        res[6]=copx[0]; res[7]=0.0f;    res[8]=copz[0];
        res[9]=fsx[0];  res[10]=fsy[0]; res[11]=fsz[0];
        res[12]=fsx[1]; res[13]=fsy[1]; res[14]=fsz[1];
        res[15]=tqx[1]; res[16]=tqy[1]; res[17]=tqz[1];
        res[18]=tqx[2]; res[19]=tqy[2]; res[20]=tqz[2];
        res[21]=copx[1]; res[22]=0.0f;  res[23]=copz[1];
        res[24]=copx[2]; res[25]=0.0f;  res[26]=copz[2];

        // wide b128 stores: sphere_forces (offset 9B) and pos (offset 45B)
        const size_t bb = (size_t)b;
        float4* sf4 = reinterpret_cast<float4*>(out +  9*sB + bb*36);
        float4* pp4 = reinterpret_cast<float4*>(out + 45*sB + bb*36);
#pragma unroll
        for (int q = 0; q < 9; ++q) {
            sf4[q] = make_float4(fl[4*q+0], fl[4*q+1], fl[4*q+2], fl[4*q+3]);
            pp4[q] = make_float4(p[4*q+0], p[4*q+1], p[4*q+2], p[4*q+3]);
        }
    }

    // reuse sm: stage the nine (B,3) outputs for fully coalesced b32 writes
    __syncthreads();   // WAR: everyone done reading the pos buffer
#pragma unroll
    for (int i = 0; i < 27; ++i) sm[tid*28 + i] = res[i];
    __syncthreads();

    const size_t mult[9] = {0, 3, 6, 81, 84, 87, 90, 93, 96};
    const size_t base3 = (size_t)blockIdx.x * (THREADS*3);
    const size_t limit = sB * 3;
#pragma unroll
    for (int r = 0; r < 9; ++r) {
        float* dst = out + mult[r]*sB + base3;
#pragma unroll
        for (int k = 0; k < 3; ++k) {
            const int idx = tid + k*THREADS;
            if (base3 + (size_t)idx < limit) {
                const int row  = idx / 3;
                const int comp = idx - row*3;
#if defined(CK_ASYNC_STORE)
                // gfx1250 async LDS -> global store (ASYNCcnt-tracked)
                __builtin_amdgcn_global_store_async_from_lds_b32(
                    (GlobalInt*)(dst + idx),
                    (SharedInt*)(&sm[row*28 + r*3 + comp]),
                    0, 0);
#else
                dst[idx] = sm[row*28 + r*3 + comp];
#endif
            }
        }
    }
#if defined(CK_ASYNC_STORE) && defined(CK_WAIT_ASYNC)
    __builtin_amdgcn_s_wait_asynccnt(0);
#endif
}

} // anonymous namespace

extern "C" void kernel_launch(void* const* d_in, const int* in_sizes, int n_in,
                              void* d_out, int out_size, void* d_ws, size_t ws_size,
                              hipStream_t stream)
{
    const float* joints = reinterpret_cast<const float*>(d_in[0]); // (B,24,3)
    const float* ori    = reinterpret_cast<const float*>(d_in[1]); // (B,24,3,3)
    float* out = reinterpret_cast<float*>(d_out);                  // 99*B floats
    const int B = in_sizes[0] / 72;
    const int grid = (B + THREADS - 1) / THREADS;
    contact_kernel<<<dim3(grid), dim3(THREADS), 0, stream>>>(joints, ori, out, B);
    (void)n_in; (void)out_size; (void)d_ws; (void)ws_size;
}